// rca_model_19653770347033
// MI455X (gfx1250) — compile-verified
//
#include <hip/hip_runtime.h>
#include <math.h>

// ---------------------------------------------------------------------------
// Problem: B=8, C=2, D=64, H=128, W=128, P=10 protos, Q=10 queue rows.
//   sim[b,v,p] = dot(x[b,:,v], protos[p,:])  -> argmax_p -> histogram counts
//   pool[b,c]  = sum_p count[b,p]*protos[p,c]   (the 's' scalar cancels, label unused)
//   then a tiny InfoNCE-style epilogue -> single scalar.
// Memory floor: read x once = 64 MiB -> ~2.9us @ 23.3 TB/s. label (16 MiB) skipped.
// The [Nvox,2]x[2,10] contraction is done with V_WMMA_F32_16X16X4_F32 (f32, K=4>=C=2).
// ---------------------------------------------------------------------------

typedef float v2f __attribute__((ext_vector_type(2)));
typedef float v8f __attribute__((ext_vector_type(8)));

#define SPATIAL (64u * 128u * 128u)   // voxels per sample = 1,048,576
#define NTILES  (SPATIAL / 16u)       // 16-voxel tiles per sample
#define NPROTO  10
#define NQ      10
#define BATCH   8

// ---------------------------------------------------------------------------
__global__ void zero_counts_kernel(unsigned* __restrict__ cnt) {
    if (threadIdx.x < BATCH * 16) cnt[threadIdx.x] = 0u;
}

// ---------------------------------------------------------------------------
// WMMA layout assumptions (CDNA5 ISA 7.12.2, f32 16x16x4):
//   A (16x4, MxK): lanes 0-15 hold row M=lane {K0,K1} in 2 VGPRs; lanes 16-31 hold {K2,K3}.
//   B (4x16, KxN): lanes 0-15 hold col N=lane {K0,K1};          lanes 16-31 hold {K2,K3}.
//   D (16x16)   : lane n in 0-15 -> col N=n, VGPR r = row M=r; lanes 16-31 -> M=8+r.
// We put protos in A (rows = proto index, K = channel, zero-padded) and a tile of
// 16 voxels in B (cols = voxels). Then lane n's D regs are sims of protos 0..7 for
// voxel n (lo half) / protos 8..15 (hi half). protos 8,9 cross over via shfl_xor(16).
__global__ __launch_bounds__(256) void
argmax_hist_kernel(const float* __restrict__ x,
                   const float* __restrict__ protos,
                   unsigned* __restrict__ gcnt) {
    const unsigned b    = blockIdx.y;
    const unsigned lane = threadIdx.x & 31u;
    const unsigned wave = threadIdx.x >> 5;

    __shared__ unsigned hist[16];
    if (threadIdx.x < 16) hist[threadIdx.x] = 0u;
    __syncthreads();

    // A-matrix: prototypes, loop invariant. lanes>=16 and rows>=NPROTO are zero.
    v2f A = {0.0f, 0.0f};
    if (lane < (unsigned)NPROTO) {            // only lanes 0..9 carry data
        A.x = protos[lane * 2 + 0];           // K0 = channel 0
        A.y = protos[lane * 2 + 1];           // K1 = channel 1
    }

    const float* __restrict__ x0 = x + (size_t)b * 2u * SPATIAL; // channel 0
    const float* __restrict__ x1 = x0 + SPATIAL;                 // channel 1

    const unsigned wavesPerBlock = blockDim.x >> 5;
    const unsigned gridWaves     = gridDim.x * wavesPerBlock;
    const unsigned waveId        = blockIdx.x * wavesPerBlock + wave;

    for (unsigned t = waveId; t < NTILES; t += gridWaves) {
        const unsigned vox = t * 16u + (lane & 15u);

        v2f Bv = {0.0f, 0.0f};
        if (lane < 16u) {                     // hi lanes = K rows 2,3 = zero pad
            Bv.x = x0[vox];
            Bv.y = x1[vox];
        }

        v8f C = {};
        // (neg_a, A, neg_b, B, c_mod, C, reuse_a, reuse_b)
        C = __builtin_amdgcn_wmma_f32_16x16x4_f32(false, A, false, Bv,
                                                  (short)0, C, false, false);

        // Within-lane argmax over protos 0..7 (valid for lo lanes).
        // Strict '>' ascending keeps FIRST max (jnp.argmax tie-break).
        float bestV = C[0];
        int   bestI = 0;
#pragma unroll
        for (int r = 1; r < 8; ++r) {
            if (C[r] > bestV) { bestV = C[r]; bestI = r; }
        }
        // protos 8,9 live in the hi-half partner lane's C[0],C[1].
        const float v8p = __shfl_xor(C[0], 16, 32);
        const float v9p = __shfl_xor(C[1], 16, 32);
        if (lane < 16u) {
            if (v8p > bestV) { bestV = v8p; bestI = 8; }
            if (v9p > bestV) { bestI = 9; }
            atomicAdd(&hist[bestI], 1u);      // integer LDS atomic: deterministic
        }
    }

    __syncthreads();
    if (threadIdx.x < 16)
        atomicAdd(&gcnt[b * 16 + threadIdx.x], hist[threadIdx.x]);
}

// ---------------------------------------------------------------------------
// Tiny epilogue: pool -> normalize -> logits -> loss. One block of 32 threads.
__global__ void epilogue_kernel(const unsigned* __restrict__ cnt,
                                const float* __restrict__ protos,
                                const float* __restrict__ q0,
                                const float* __restrict__ q1,
                                float* __restrict__ out) {
    __shared__ float fneg[NQ][2];
    __shared__ float fpos[NQ][2];
    __shared__ float lbuf[BATCH];
    const int t = threadIdx.x;

    if (t < NQ) {
        float a = q0[t * 2 + 0], c = q0[t * 2 + 1];
        float n = fmaxf(sqrtf(a * a + c * c), 1e-12f);
        fneg[t][0] = a / n; fneg[t][1] = c / n;
        a = q1[t * 2 + 0]; c = q1[t * 2 + 1];
        n = fmaxf(sqrtf(a * a + c * c), 1e-12f);
        fpos[t][0] = a / n; fpos[t][1] = c / n;
    }
    __syncthreads();

    if (t < BATCH) {
        // pool[b,:] = sum_p count[b,p] * protos[p,:]   (the 's' factor cancels)
        float p0 = 0.0f, p1 = 0.0f;
        for (int p = 0; p < NPROTO; ++p) {
            const float c = (float)cnt[t * 16 + p];
            p0 += c * protos[p * 2 + 0];
            p1 += c * protos[p * 2 + 1];
        }
        const float n = fmaxf(sqrtf(p0 * p0 + p1 * p1), 1e-12f);
        p0 /= n; p1 /= n;

        float ln[NQ], maxl = -INFINITY;
        for (int q = 0; q < NQ; ++q) {
            ln[q] = (p0 * fneg[q][0] + p1 * fneg[q][1]) / 0.07f;
            maxl  = fmaxf(maxl, ln[q]);
        }
        float lneg = 0.0f;
        for (int q = 0; q < NQ; ++q) lneg += expf(ln[q] - maxl);

        float loss = 0.0f;
        for (int q = 0; q < NQ; ++q) {
            const float lp  = (p0 * fpos[q][0] + p1 * fpos[q][1]) / 0.07f - maxl;
            const float den = fmaxf(lneg + expf(lp), 1e-4f);  // clip(min=1e-4)
            loss += -(lp - logf(den));
        }
        lbuf[t] = loss / (float)NQ;
    }
    __syncthreads();

    if (t == 0) {
        float s = 0.0f;
        for (int i = 0; i < BATCH; ++i) s += lbuf[i];
        out[0] = s / (float)BATCH;
    }
}

// ---------------------------------------------------------------------------
extern "C" void kernel_launch(void* const* d_in, const int* in_sizes, int n_in,
                              void* d_out, int out_size, void* d_ws, size_t ws_size,
                              hipStream_t stream) {
    const float* x      = (const float*)d_in[0];
    // d_in[1] = label: provably unused (scalar s cancels in pool = proto*s.sum()/s)
    const float* protos = (const float*)d_in[2];
    const float* q0     = (const float*)d_in[3];
    const float* q1     = (const float*)d_in[4];
    float* out          = (float*)d_out;
    unsigned* cnt       = (unsigned*)d_ws;   // BATCH*16 counters = 512 bytes

    zero_counts_kernel<<<1, 128, 0, stream>>>(cnt);

    dim3 grid(256, BATCH, 1);                // 2048 blocks, 8 wave32 per block
    argmax_hist_kernel<<<grid, 256, 0, stream>>>(x, protos, cnt);

    epilogue_kernel<<<1, 32, 0, stream>>>(cnt, protos, q0, q1, out);
}